// QueriedAttn2D_26869315403839
// MI455X (gfx1250) — compile-verified
//
#include <hip/hip_runtime.h>
#include <math.h>

typedef __attribute__((ext_vector_type(2))) float v2f;
typedef __attribute__((ext_vector_type(8))) float v8f;

#define D_MODEL 256
#define D_HEAD  64
#define N_HEAD  8
#define DK      512    // N_HEAD*D_HEAD
#define T       4096   // 64*64
#define NB      32

// workspace layout (bytes)
#define OFF_U    0u          // [NB][16][256] f32, rows 8..15 zero, pre-scaled
#define OFF_CC   524288u     // [NB][8] f32, pre-scaled
#define OFF_LOG  525312u     // [NB][8][T] f32
#define OFF_P    4719616u    // [NB][T][16] f32 (cols 8..15 zero)
#define OFF_YP   13108224u   // [4][NB][256][16] f32 partials

// ---------------- Kernel 1: q, U = scale*(q@Wkv_k), cconst ----------------
__global__ void k1_prep(const float* __restrict__ query, const float* __restrict__ Wq,
                        const float* __restrict__ bq, const float* __restrict__ Wkv,
                        const float* __restrict__ bkv, float* __restrict__ U,
                        float* __restrict__ cc) {
  __shared__ float qv[D_MODEL];
  __shared__ float qs[DK];
  const int b = blockIdx.x, t = threadIdx.x;
  if (t < D_MODEL) qv[t] = query[b * D_MODEL + t];
  __syncthreads();
  {  // q[b][t], t = 0..511
    float acc = bq[t];
    const float* w = Wq + (size_t)t * D_MODEL;
    for (int c = 0; c < D_MODEL; ++c) acc += qv[c] * w[c];
    qs[t] = acc;
  }
  __syncthreads();
  const float scale = 0.125f;  // 1/sqrt(D_HEAD)
  // U[b][n][c] = scale * sum_d q[n*64+d] * Wkv[n*64+d][c]
  for (int i = t; i < N_HEAD * D_MODEL; i += blockDim.x) {
    const int n = i >> 8, c = i & 255;
    float acc = 0.f;
    const float* qn = qs + n * D_HEAD;
    const float* w  = Wkv + (size_t)(n * D_HEAD) * D_MODEL + c;
    #pragma unroll 4
    for (int d = 0; d < D_HEAD; ++d) acc += qn[d] * w[(size_t)d * D_MODEL];
    U[((size_t)b * 16 + n) * D_MODEL + c] = acc * scale;
  }
  // zero-pad head rows 8..15 (so WMMA M-padding yields zeros)
  for (int i = t; i < 8 * D_MODEL; i += blockDim.x)
    U[((size_t)b * 16 + 8) * D_MODEL + i] = 0.f;
  if (t < N_HEAD) {
    float acc = 0.f;
    for (int d = 0; d < D_HEAD; ++d) acc += qs[t * D_HEAD + d] * bkv[t * D_HEAD + d];
    cc[b * N_HEAD + t] = acc * scale;
  }
}

// ---------------- Kernel 2: logits = U_pad(16x256) x X(256x4096) via WMMA f32 16x16x4 ----------------
__global__ void __launch_bounds__(256) k2_logits(const float* __restrict__ x,
                                                 const float* __restrict__ U,
                                                 const float* __restrict__ cc,
                                                 float* __restrict__ logits) {
  __shared__ float Ul[16 * 260];  // stride 260 words: lanes m=0..15 hit distinct banks
  __shared__ float ccl[8];
  const int b = blockIdx.x >> 5;       // 32 j-blocks per batch
  const int jblk = blockIdx.x & 31;    // each block covers 128 columns
  const int t = threadIdx.x;
  for (int i = t; i < 16 * D_MODEL; i += 256)
    Ul[(i >> 8) * 260 + (i & 255)] = U[(size_t)b * 16 * D_MODEL + i];
  if (t < 8) ccl[t] = cc[b * 8 + t];
  __syncthreads();
  const int lane = t & 31, wave = t >> 5;
  const int m = lane & 15, hi = lane >> 4;
  const int j = jblk * 128 + wave * 16 + m;  // B: N = lane%16
  // B lanes 16-31 carry K rows c+2,c+3 -> fold 2*hi rows into base
  const float* xb = x + (size_t)b * D_MODEL * T + (size_t)(2 * hi) * T + j;
  const float* arow = Ul + m * 260 + 2 * hi;  // A: lanes 16-31 carry K=c+2,c+3
  v8f acc = {};
  #pragma unroll 8
  for (int s = 0; s < 64; ++s) {
    const int c = 4 * s;
    v2f a = *(const v2f*)(arow + c);        // ds_load_b64: U[m][c+2hi], U[m][c+1+2hi]
    v2f bv;
    bv.x = xb[(size_t)c * T];               // x[c+2hi][j]
    bv.y = xb[(size_t)(c + 1) * T];         // x[c+1+2hi][j]
    acc = __builtin_amdgcn_wmma_f32_16x16x4_f32(false, a, false, bv,
                                                (short)0, acc, false, false);
  }
  // D: VGPR i -> row i (lanes 0-15) / row i+8 (lanes 16-31); rows 8..15 are padding
  if (hi == 0) {
    float* lp = logits + (size_t)b * 8 * T + (jblk * 128 + wave * 16 + lane);
    #pragma unroll
    for (int i = 0; i < 8; ++i) lp[(size_t)i * T] = acc[i] + ccl[i];
  }
}

// ---------------- Kernel 3: softmax over t, write padded P[b][j][16] ----------------
__global__ void k3_softmax(const float* __restrict__ logits, float* __restrict__ P) {
  __shared__ float red[256];
  const int b = blockIdx.x >> 3, n = blockIdx.x & 7;
  const int t = threadIdx.x;
  const float* lp = logits + ((size_t)b * 8 + n) * T;
  float v[16];
  float mx = -INFINITY;
  #pragma unroll
  for (int i = 0; i < 16; ++i) { v[i] = lp[t + i * 256]; mx = fmaxf(mx, v[i]); }
  red[t] = mx; __syncthreads();
  for (int s = 128; s > 0; s >>= 1) {
    if (t < s) red[t] = fmaxf(red[t], red[t + s]);
    __syncthreads();
  }
  mx = red[0]; __syncthreads();
  float sum = 0.f;
  #pragma unroll
  for (int i = 0; i < 16; ++i) { v[i] = __expf(v[i] - mx); sum += v[i]; }
  red[t] = sum; __syncthreads();
  for (int s = 128; s > 0; s >>= 1) {
    if (t < s) red[t] += red[t + s];
    __syncthreads();
  }
  const float inv = 1.f / red[0];
  float* pb = P + (size_t)b * T * 16;
  #pragma unroll
  for (int i = 0; i < 16; ++i) {
    const int j = t + i * 256;
    pb[(size_t)j * 16 + n] = v[i] * inv;
    if (n == 0) {
      #pragma unroll
      for (int z = 8; z < 16; ++z) pb[(size_t)j * 16 + z] = 0.f;  // pad columns
    }
  }
}

// ---------------- Kernel 4: y = X(256x4096) x P_pad(4096x16) via WMMA, 4-way K split ----------------
__global__ void __launch_bounds__(256) k4_y(const float* __restrict__ x,
                                            const float* __restrict__ P,
                                            float* __restrict__ ypart) {
  const int t = threadIdx.x;
  const int lane = t & 31, wave = t >> 5;
  const int bid = blockIdx.x;          // 256 blocks = 32b x 4kc x 2ctg
  const int b = bid >> 3;
  const int r = bid & 7;
  const int kc = r & 3;
  const int ctg = r >> 2;
  const int c0 = (ctg * 8 + wave) * 16;
  const int n = lane & 15, hi = lane >> 4;
  const int k0 = kc * 1024;
  // A: row m = c0 + lane%16, K pair at k+2*hi -> contiguous float2 along j
  const float* xa = x + ((size_t)b * D_MODEL + (c0 + n)) * T + k0 + 2 * hi;
  // B: column n = lane%16, K rows (k+2hi, k+1+2hi)
  const float* pb = P + ((size_t)b * T + k0 + 2 * hi) * 16 + n;
  v8f acc = {};
  #pragma unroll 8
  for (int s = 0; s < 256; ++s) {
    const int k = 4 * s;
    v2f a = *(const v2f*)(xa + k);          // global_load_b64
    v2f bv;
    bv.x = pb[(size_t)k * 16];
    bv.y = pb[(size_t)(k + 1) * 16];
    acc = __builtin_amdgcn_wmma_f32_16x16x4_f32(false, a, false, bv,
                                                (short)0, acc, false, false);
  }
  float* yp = ypart + ((size_t)kc * NB + b) * D_MODEL * 16;
  #pragma unroll
  for (int i = 0; i < 8; ++i) {
    const int c = c0 + i + 8 * hi;
    yp[(size_t)c * 16 + n] = acc[i];
  }
}

// ---------------- Kernel 5: reduce partials, out = Wkv_v@y + bkv_v, res = relu(out@Wfc.T + bfc) ----------------
__global__ void k5_out(const float* __restrict__ Wkv, const float* __restrict__ bkv,
                       const float* __restrict__ Wfc, const float* __restrict__ bfc,
                       const float* __restrict__ ypart, float* __restrict__ out) {
  __shared__ float yl[D_MODEL][N_HEAD];
  __shared__ float ol[DK];
  const int b = blockIdx.x, t = threadIdx.x;
  for (int i = t; i < D_MODEL * N_HEAD; i += blockDim.x) {
    const int c = i >> 3, n = i & 7;
    float s = 0.f;
    #pragma unroll
    for (int kc = 0; kc < 4; ++kc)
      s += ypart[(((size_t)kc * NB + b) * D_MODEL + c) * 16 + n];
    yl[c][n] = s;
  }
  __syncthreads();
  {  // out[t] = bkv_v[t] + Wkv_v[t]·y[:, t/64]
    const int n = t >> 6;
    const float* w = Wkv + (size_t)(DK + t) * D_MODEL;
    float acc = bkv[DK + t];
    for (int c = 0; c < D_MODEL; ++c) acc += w[c] * yl[c][n];
    ol[t] = acc;
  }
  __syncthreads();
  if (t < D_MODEL) {
    const float* w = Wfc + (size_t)t * DK;
    float acc = bfc[t];
    for (int k = 0; k < DK; ++k) acc += w[k] * ol[k];
    out[(size_t)b * D_MODEL + t] = fmaxf(acc, 0.f);
  }
}

extern "C" void kernel_launch(void* const* d_in, const int* in_sizes, int n_in,
                              void* d_out, int out_size, void* d_ws, size_t ws_size,
                              hipStream_t stream) {
  const float* x     = (const float*)d_in[0];
  const float* query = (const float*)d_in[1];
  const float* Wkv   = (const float*)d_in[2];
  const float* bkv   = (const float*)d_in[3];
  const float* Wq    = (const float*)d_in[4];
  const float* bq    = (const float*)d_in[5];
  const float* Wfc   = (const float*)d_in[6];
  const float* bfc   = (const float*)d_in[7];
  char* ws = (char*)d_ws;
  float* U  = (float*)(ws + OFF_U);
  float* cc = (float*)(ws + OFF_CC);
  float* lg = (float*)(ws + OFF_LOG);
  float* P  = (float*)(ws + OFF_P);
  float* yp = (float*)(ws + OFF_YP);

  k1_prep   <<<NB,   512, 0, stream>>>(query, Wq, bq, Wkv, bkv, U, cc);
  k2_logits <<<1024, 256, 0, stream>>>(x, U, cc, lg);
  k3_softmax<<<256,  256, 0, stream>>>(lg, P);
  k4_y      <<<256,  256, 0, stream>>>(x, P, yp);
  k5_out    <<<NB,   512, 0, stream>>>(Wkv, bkv, Wfc, bfc, yp, (float*)d_out);
}